// GraphGCN_13718125543734
// MI455X (gfx1250) — compile-verified
//
#include <hip/hip_runtime.h>
#include <stdint.h>

#define D        64          // feature dim
#define D4       16          // float4 per row
#define THREADS  256
#define TILE     256         // edges staged per block iteration
#define LPE      16          // lanes per edge (16 lanes x float4 = 256B row)

// ---- gfx1250 async-to-LDS path (guarded so compile always succeeds) -------
#if defined(__AMDGCN__) && defined(__gfx1250__) && \
    __has_builtin(__builtin_amdgcn_global_load_async_to_lds_b64)
#define HAVE_ASYNC_LDS 1
typedef int v2i __attribute__((vector_size(8)));
typedef __attribute__((address_space(1))) v2i* gv2i_p;  // global v2i*
typedef __attribute__((address_space(3))) v2i* lv2i_p;  // LDS v2i*
#else
#define HAVE_ASYNC_LDS 0
#endif

// Non-returning, relaxed, agent-scope f32 atomic add -> global_atomic_add_f32
__device__ __forceinline__ void atomAddF(float* p, float v) {
    __hip_atomic_fetch_add(p, v, __ATOMIC_RELAXED, __HIP_MEMORY_SCOPE_AGENT);
}

__device__ __forceinline__ void waitAsyncLE0() {
#if HAVE_ASYNC_LDS
#if __has_builtin(__builtin_amdgcn_s_wait_asynccnt)
    __builtin_amdgcn_s_wait_asynccnt(0);
#else
    asm volatile("s_wait_asynccnt 0" ::: "memory");
#endif
#endif
}

__device__ __forceinline__ void waitAsyncLE2() {
#if HAVE_ASYNC_LDS
#if __has_builtin(__builtin_amdgcn_s_wait_asynccnt)
    __builtin_amdgcn_s_wait_asynccnt(2);
#else
    asm volatile("s_wait_asynccnt 2" ::: "memory");
#endif
#endif
}

// FIRST function in the module so the disasm snippet exposes the atomic
// lowering (expect a single non-returning global_atomic_add_f32).
__global__ void degree_kernel(const long long* __restrict__ dst,
                              float* __restrict__ deg, int E) {
    int i = blockIdx.x * blockDim.x + threadIdx.x;
    int stride = gridDim.x * blockDim.x;
    for (; i < E; i += stride) {
        atomAddF(&deg[(int)dst[i]], 1.0f);
    }
}

// Gather features[src] rows and atomically accumulate into sum[dst].
// Edge indices stream into LDS via the gfx1250 async DMA path with DOUBLE
// BUFFERING: tile t+gridDim is staged while tile t is processed; the wave
// waits only until its two older async loads have landed (asynccnt <= 2).
__global__ void scatter_kernel(const float4* __restrict__ feat,
                               float* __restrict__ sum,
                               const long long* __restrict__ src,
                               const long long* __restrict__ dst,
                               int E) {
    __shared__ long long s_src[2][TILE];
    __shared__ long long s_dst[2][TILE];
    const int tid = threadIdx.x;
    const int ntiles = (E + TILE - 1) / TILE;
    int cur = 0;

#if HAVE_ASYNC_LDS
#define STAGE(buf, gidx)                                                      \
    do {                                                                      \
        __builtin_amdgcn_global_load_async_to_lds_b64(                        \
            (gv2i_p)(src + (gidx)), (lv2i_p)&s_src[(buf)][tid], 0, 0);        \
        __builtin_amdgcn_global_load_async_to_lds_b64(                        \
            (gv2i_p)(dst + (gidx)), (lv2i_p)&s_dst[(buf)][tid], 0, 0);        \
    } while (0)
#else
#define STAGE(buf, gidx)                                                      \
    do {                                                                      \
        s_src[(buf)][tid] = src[(gidx)];                                      \
        s_dst[(buf)][tid] = dst[(gidx)];                                      \
    } while (0)
#endif

    // Prologue: stage this block's first tile.
    const int t0 = blockIdx.x;
    if (t0 < ntiles) {
        const int base = t0 * TILE;
        const int cnt  = (E - base) < TILE ? (E - base) : TILE;
        if (tid < cnt) STAGE(cur, base + tid);
    }

    for (int tile = t0; tile < ntiles; tile += gridDim.x) {
        const int base = tile * TILE;
        const int cnt  = (E - base) < TILE ? (E - base) : TILE;

        // Stage the NEXT tile into the other buffer, then drain only the
        // two older async loads so the new ones overlap with processing.
        const int ntile = tile + gridDim.x;
        if (ntile < ntiles) {
            const int nbase = ntile * TILE;
            const int ncnt  = (E - nbase) < TILE ? (E - nbase) : TILE;
            if (tid < ncnt) STAGE(cur ^ 1, nbase + tid);
            // Per-wave: only waves that actually issued the stage (any lane
            // active) have 4 outstanding; others must drain fully.
            if ((tid & ~31) < ncnt) waitAsyncLE2(); else waitAsyncLE0();
        } else {
            waitAsyncLE0();
        }
        __syncthreads();

        // Warm the cache for all staged rows (256B row = 2 x 128B lines).
        if (tid < cnt) {
            const float4* row = &feat[(int)s_src[cur][tid] * D4];
            __builtin_prefetch(row, 0, 1);
            __builtin_prefetch(row + 8, 0, 1);
        }

        const int group = tid / LPE;   // which edge within the tile slice
        const int lane  = tid % LPE;   // which float4 of the 64-float row
        for (int e = group; e < cnt; e += THREADS / LPE) {
            const int s = (int)s_src[cur][e];
            const int d = (int)s_dst[cur][e];
            float4 v = feat[(size_t)s * D4 + lane];
            float* out = sum + (size_t)d * D + lane * 4;
            atomAddF(out + 0, v.x);
            atomAddF(out + 1, v.y);
            atomAddF(out + 2, v.z);
            atomAddF(out + 3, v.w);
        }
        __syncthreads();   // buffer may be overwritten next iteration
        cur ^= 1;
    }
#undef STAGE
}

__global__ void zero_kernel(float* __restrict__ p, int n) {
    int i = blockIdx.x * blockDim.x + threadIdx.x;
    int stride = gridDim.x * blockDim.x;
    for (; i < n; i += stride) p[i] = 0.0f;
}

__global__ void divide_kernel(const float4* __restrict__ sum,
                              const float* __restrict__ deg,
                              float4* __restrict__ x, int n_nodes) {
    int i = blockIdx.x * blockDim.x + threadIdx.x;
    const int total = n_nodes * D4;
    int stride = gridDim.x * blockDim.x;
    for (; i < total; i += stride) {
        const int node = i / D4;
        const float invd = 1.0f / fmaxf(deg[node], 1.0f);
        float4 v = sum[i];
        v.x *= invd; v.y *= invd; v.z *= invd; v.w *= invd;
        x[i] = v;
    }
}

// out = 0.5 * (x + sum/max(deg,1)), with x already resident in out
__global__ void final_kernel(float4* __restrict__ out,
                             const float4* __restrict__ sum,
                             const float* __restrict__ deg, int n_nodes) {
    int i = blockIdx.x * blockDim.x + threadIdx.x;
    const int total = n_nodes * D4;
    int stride = gridDim.x * blockDim.x;
    for (; i < total; i += stride) {
        const int node = i / D4;
        const float invd = 1.0f / fmaxf(deg[node], 1.0f);
        float4 s = sum[i];
        float4 x = out[i];
        float4 r;
        r.x = 0.5f * (x.x + s.x * invd);
        r.y = 0.5f * (x.y + s.y * invd);
        r.z = 0.5f * (x.z + s.z * invd);
        r.w = 0.5f * (x.w + s.w * invd);
        out[i] = r;
    }
}

extern "C" void kernel_launch(void* const* d_in, const int* in_sizes, int n_in,
                              void* d_out, int out_size, void* d_ws, size_t ws_size,
                              hipStream_t stream) {
    const float*     feat = (const float*)d_in[0];
    const long long* src  = (const long long*)d_in[1];
    const long long* dst  = (const long long*)d_in[2];
    const int E       = in_sizes[1];
    const int n_nodes = in_sizes[0] / D;

    float* sum = (float*)d_ws;                     // n_nodes * D floats (25.6 MB)
    float* deg = sum + (size_t)n_nodes * D;        // n_nodes floats
    float* out = (float*)d_out;

    const int sumN   = n_nodes * D;
    const int ntiles = (E + TILE - 1) / TILE;
    const int sgrid  = ntiles < 8192 ? ntiles : 8192;

    // 1) zero accumulator + degrees
    zero_kernel<<<2048, THREADS, 0, stream>>>(sum, sumN + n_nodes);
    // 2) degrees (shared by both layers)
    degree_kernel<<<2048, THREADS, 0, stream>>>(dst, deg, E);
    // 3) layer-1 scatter-add
    scatter_kernel<<<sgrid, THREADS, 0, stream>>>((const float4*)feat, sum, src, dst, E);
    // 4) x = sum / max(deg,1)  -> stored in d_out
    divide_kernel<<<2048, THREADS, 0, stream>>>((const float4*)sum, deg, (float4*)out, n_nodes);
    // 5) re-zero accumulator
    zero_kernel<<<2048, THREADS, 0, stream>>>(sum, sumN);
    // 6) layer-2 scatter-add reading x from d_out
    scatter_kernel<<<sgrid, THREADS, 0, stream>>>((const float4*)out, sum, src, dst, E);
    // 7) out = 0.5 * (x + sum/max(deg,1))
    final_kernel<<<2048, THREADS, 0, stream>>>((float4*)out, (const float4*)sum, deg, n_nodes);
}